// ChamferDistance_5987184411285
// MI455X (gfx1250) — compile-verified
//
#include <hip/hip_runtime.h>
#include <hip/hip_bf16.h>

typedef float v2f __attribute__((ext_vector_type(2)));
typedef float v8f __attribute__((ext_vector_type(8)));

#define TM 256                 // reference points staged in LDS per chunk
#define WAVES 8
#define ROWS_PER_BLOCK (WAVES * 16)

// For each point in apts[batch], min squared distance to all points in bpts[batch].
// Distance tile computed entirely inside one V_WMMA_F32_16X16X4_F32:
//   A row  = ( x,  y,  z, 1 )          (query point)
//   B col  = (-2x, -2y, -2z, ||b||^2)  (reference point, pre-transformed in LDS)
//   C      =  ||a||^2  (row-broadcast; loop-invariant across the M sweep)
//   D[i][j] = ||a_i||^2 - 2 a_i.b_j + ||b_j||^2
__global__ __launch_bounds__(256) void chamfer_min_kernel(
    const float* __restrict__ apts,   // [B, nA, 3]
    const float* __restrict__ bpts,   // [B, nB, 3]
    float* __restrict__ out,          // [B, nA]
    int nA, int nB)
{
    __shared__ float4 smem[TM];       // (-2x,-2y,-2z,||b||^2) records, 16B each

    const int batch  = blockIdx.y;
    const int tid    = threadIdx.x;
    const int lane   = tid & 31;
    const int wave   = tid >> 5;
    const int half   = lane >> 4;     // 0: lanes 0-15, 1: lanes 16-31
    const int lane16 = lane & 15;

    const float* abase = apts + (size_t)batch * nA * 3;
    const float* bbase = bpts + (size_t)batch * nB * 3;

    // ---- this wave's 16 query points; build A operand ----
    const int rowBase = blockIdx.x * ROWS_PER_BLOCK + wave * 16;
    const int arow    = rowBase + lane16;
    const float ax = abase[(size_t)arow * 3 + 0];
    const float ay = abase[(size_t)arow * 3 + 1];
    const float az = abase[(size_t)arow * 3 + 2];
    const float sqa = ax * ax + ay * ay + az * az;

    v2f aop;                           // A 16x4 layout: lanes0-15=(K0,K1), lanes16-31=(K2,K3)
    aop.x = half ? az   : ax;
    aop.y = half ? 1.0f : ay;

    // ---- C operand: c[r] = ||a_row||^2 with row = r + 8*half (loop-invariant) ----
    v8f cop;
#pragma unroll
    for (int r = 0; r < 8; ++r)
        cop[r] = __shfl(sqa, r + (half ? 8 : 0), 32);

    v8f acc;
#pragma unroll
    for (int r = 0; r < 8; ++r)
        acc[r] = __builtin_huge_valf();

    for (int mt = 0; mt < nB; mt += TM) {
        // cooperative staging: one transformed reference point per thread
        {
            const int p = mt + tid;
            const float* bp = bbase + (size_t)p * 3;
            const float bx = bp[0], by = bp[1], bz = bp[2];
            const float sqb = bx * bx + by * by + bz * bz;
            smem[tid] = make_float4(-2.0f * bx, -2.0f * by, -2.0f * bz, sqb);
            if (mt + TM < nB)
                __builtin_prefetch(bbase + (size_t)(p + TM) * 3, 0, 0);
        }
        __syncthreads();

        const float2* recs = (const float2*)smem;
#pragma unroll 4
        for (int jt = 0; jt < TM / 16; ++jt) {
            // B operand: column n = jt*16 + lane16; halves pick (K0,K1)/(K2,K3)
            const float2 bv = recs[(jt * 16 + lane16) * 2 + half];
            v2f bop;
            bop.x = bv.x;
            bop.y = bv.y;

            v8f d = __builtin_amdgcn_wmma_f32_16x16x4_f32(
                /*neg_a=*/false, aop, /*neg_b=*/false, bop,
                /*c_mod=*/(short)0, cop, /*reuse_a=*/false, /*reuse_b=*/false);

#pragma unroll
            for (int r = 0; r < 8; ++r)
                acc[r] = fminf(acc[r], d[r]);
        }
        __syncthreads();
    }

    // ---- min across the 16 columns held in this lane half ----
#pragma unroll
    for (int m = 1; m < 16; m <<= 1) {
#pragma unroll
        for (int r = 0; r < 8; ++r)
            acc[r] = fminf(acc[r], __shfl_xor(acc[r], m, 32));
    }

    // D layout: VGPR r <-> row r (lanes 0-15) / row r+8 (lanes 16-31)
    float* orow = out + (size_t)batch * nA + rowBase;
    if (lane == 0) {
#pragma unroll
        for (int r = 0; r < 8; ++r)
            orow[r] = fmaxf(acc[r], 0.0f);
    } else if (lane == 16) {
#pragma unroll
        for (int r = 0; r < 8; ++r)
            orow[8 + r] = fmaxf(acc[r], 0.0f);
    }
}

extern "C" void kernel_launch(void* const* d_in, const int* in_sizes, int n_in,
                              void* d_out, int out_size, void* d_ws, size_t ws_size,
                              hipStream_t stream) {
    const float* xyz1 = (const float*)d_in[0];   // [B, N, 3]
    const float* xyz2 = (const float*)d_in[1];   // [B, M, 3]
    float* out = (float*)d_out;                  // dist1 [B,N] then dist2 [B,M]

    const int B = 4;
    const int N = in_sizes[0] / (B * 3);
    const int M = in_sizes[1] / (B * 3);

    dim3 block(256);
    dim3 g1((N + ROWS_PER_BLOCK - 1) / ROWS_PER_BLOCK, B);
    dim3 g2((M + ROWS_PER_BLOCK - 1) / ROWS_PER_BLOCK, B);

    // dist1: for each xyz1 point, nearest xyz2 point
    chamfer_min_kernel<<<g1, block, 0, stream>>>(xyz1, xyz2, out, N, M);
    // dist2: for each xyz2 point, nearest xyz1 point
    chamfer_min_kernel<<<g2, block, 0, stream>>>(xyz2, xyz1, out + (size_t)B * N, M, N);
}